// GBottleneck_66048007078519
// MI455X (gfx1250) — compile-verified
//
#include <hip/hip_runtime.h>

typedef __attribute__((ext_vector_type(2))) float v2f;
typedef __attribute__((ext_vector_type(8))) float v8f;

#define NEG_SLOPE 0.01f

// ---------------- degree / norm ----------------
__global__ void k_norm_init(float* __restrict__ norm, int n) {
  int i = blockIdx.x * blockDim.x + threadIdx.x;
  if (i < n) norm[i] = 1.0f;  // self-loop folded into degree
}
__global__ void k_norm_count(const int* __restrict__ dst, float* __restrict__ norm, int e) {
  int i = blockIdx.x * blockDim.x + threadIdx.x;
  if (i < e) atomicAdd(&norm[dst[i]], 1.0f);
}
__global__ void k_norm_rsqrt(float* __restrict__ norm, int n) {
  int i = blockIdx.x * blockDim.x + threadIdx.x;
  if (i < n) norm[i] = rsqrtf(norm[i]);
}

// ---------------- GEMM with f32 WMMA ----------------
// Computes S[i,:] = (X[i,:] @ W) * norm[i]  and AGG = S (self-loop init).
// Block: 256 threads = 8 waves, 128 rows per block (16 rows per wave).
// LDS: X tile 128 x (K+4) row-major; W staged TRANSPOSED: Wt[n][k], stride K+4.
// Both A and B fragments are then contiguous K-pairs -> single ds_load_b64 each,
// bank-conflict-free with the +4 pad (bank = (4*row + ka) mod 64).
__global__ __launch_bounds__(256) void k_gemm_wmma(
    const float* __restrict__ X, const float* __restrict__ W,
    const float* __restrict__ norm,
    float* __restrict__ S, float* __restrict__ AGG,
    int nrows, int K, int Hout) {
  extern __shared__ float lds[];
  const int KSTRIDE = K + 4;
  float* Xs = lds;                    // [128][KSTRIDE]
  float* Wt = lds + 128 * KSTRIDE;    // [Hout][KSTRIDE]  (transposed W)

  const int tid = threadIdx.x;
  const int row0 = blockIdx.x * 128;

  // Stage W (K x Hout, row-major in global) transposed into LDS.
  for (int idx = tid * 4; idx < K * Hout; idx += 256 * 4) {
    int k = idx / Hout, c = idx - k * Hout;
    float4 v = *(const float4*)&W[idx];
    Wt[(c + 0) * KSTRIDE + k] = v.x;
    Wt[(c + 1) * KSTRIDE + k] = v.y;
    Wt[(c + 2) * KSTRIDE + k] = v.z;
    Wt[(c + 3) * KSTRIDE + k] = v.w;
  }
  // Stage X tile (128 x K) into LDS, zero-padded past nrows.
  for (int idx = tid * 4; idx < 128 * K; idx += 256 * 4) {
    int r = idx / K, c = idx - r * K;
    float4 v = {0.f, 0.f, 0.f, 0.f};
    int gr = row0 + r;
    if (gr < nrows) v = *(const float4*)&X[(long)gr * K + c];
    *(float4*)&Xs[r * KSTRIDE + c] = v;
  }
  __syncthreads();

  const int lane = tid & 31;
  const int wave = tid >> 5;        // row tile within block
  const int mrow = lane & 15;       // A-frag row (both half-waves cover M=0..15)
  const int kg   = lane >> 4;       // K-pair select: 0 -> K{0,1}, 1 -> K{2,3}
  const float* Xrow = &Xs[(wave * 16 + mrow) * KSTRIDE + 2 * kg];

  // norm for the 8 output rows this lane owns (C/D: VGPR r -> M = r + 8*(lane>>4))
  float nrm[8];
#pragma unroll
  for (int r = 0; r < 8; ++r) {
    int grow = row0 + wave * 16 + r + 8 * kg;
    nrm[r] = (grow < nrows) ? norm[grow] : 0.f;
  }

  for (int ct = 0; ct < (Hout >> 4); ++ct) {
    const int col = ct * 16 + mrow;
    const float* Wcol = &Wt[col * KSTRIDE + 2 * kg];
    v8f acc = {};
    for (int k = 0; k < K; k += 4) {
      // contiguous, 8B-aligned pairs -> ds_load_b64 for both fragments
      v2f a = *(const v2f*)&Xrow[k];
      v2f b = *(const v2f*)&Wcol[k];
      // D = A(16x4) * B(4x16) + C, fp32 in/out
      acc = __builtin_amdgcn_wmma_f32_16x16x4_f32(
          /*neg_a=*/false, a, /*neg_b=*/false, b,
          /*c_mod=*/(short)0, acc, /*reuse_a=*/false, /*reuse_b=*/false);
    }
#pragma unroll
    for (int r = 0; r < 8; ++r) {
      int grow = row0 + wave * 16 + r + 8 * kg;
      if (grow < nrows) {
        float sv = acc[r] * nrm[r];
        long o = (long)grow * Hout + col;
        S[o]   = sv;   // message value: xw*norm
        AGG[o] = sv;   // self-loop init (finalize multiplies by norm again)
      }
    }
  }
}

// ---------------- edge scatter: AGG[dst] += S[src] ----------------
// One float4 chunk per thread; chunks of one edge occupy consecutive lanes,
// so the 4 atomics per lane form one contiguous 512B region per edge.
// S and AGG (25.6 MB each) are L2-resident on the 192 MB L2.
__global__ void k_scatter(const int* __restrict__ src, const int* __restrict__ dst,
                          const float* __restrict__ S, float* __restrict__ AGG,
                          int e, int Hout) {
  const int chunks = Hout >> 2;
  long tid = (long)blockIdx.x * blockDim.x + threadIdx.x;
  if (tid >= (long)e * chunks) return;
  int ed = (int)(tid / chunks);
  int c  = (int)(tid - (long)ed * chunks) << 2;
  int s = src[ed], d = dst[ed];
  const float4 v = *(const float4*)&S[(long)s * Hout + c];
  float* ap = &AGG[(long)d * Hout + c];
  atomicAdd(ap + 0, v.x);
  atomicAdd(ap + 1, v.y);
  atomicAdd(ap + 2, v.z);
  atomicAdd(ap + 3, v.w);
}

// ---------------- finalize: out = act(AGG*norm + b) [+ residual] ----------------
// mode 0: leaky only; mode 1: out = (xin + leaky(v)) * 0.5; mode 2: no activation
__global__ void k_finalize(const float* __restrict__ AGG, const float* __restrict__ norm,
                           const float* __restrict__ bias, const float* __restrict__ xin,
                           float* __restrict__ out, int n, int Hout, int mode) {
  const int chunks = Hout >> 2;
  long tid = (long)blockIdx.x * blockDim.x + threadIdx.x;
  if (tid >= (long)n * chunks) return;
  int i = (int)(tid / chunks);
  int c = (int)(tid - (long)i * chunks) << 2;
  long o = (long)i * Hout + c;
  float nr = norm[i];
  float4 av = *(const float4*)&AGG[o];
  float4 bv = *(const float4*)&bias[c];
  float vx = av.x * nr + bv.x;
  float vy = av.y * nr + bv.y;
  float vz = av.z * nr + bv.z;
  float vw = av.w * nr + bv.w;
  if (mode != 2) {
    vx = vx >= 0.f ? vx : NEG_SLOPE * vx;
    vy = vy >= 0.f ? vy : NEG_SLOPE * vy;
    vz = vz >= 0.f ? vz : NEG_SLOPE * vz;
    vw = vw >= 0.f ? vw : NEG_SLOPE * vw;
  }
  if (mode == 1) {
    float4 xv = *(const float4*)&xin[o];
    vx = (xv.x + vx) * 0.5f;
    vy = (xv.y + vy) * 0.5f;
    vz = (xv.z + vz) * 0.5f;
    vw = (xv.w + vw) * 0.5f;
  }
  float4 r4 = {vx, vy, vz, vw};
  *(float4*)&out[o] = r4;
}

// ---------------- orchestration ----------------
extern "C" void kernel_launch(void* const* d_in, const int* in_sizes, int n_in,
                              void* d_out, int out_size, void* d_ws, size_t ws_size,
                              hipStream_t stream) {
  (void)n_in; (void)out_size; (void)ws_size;
  const float* inputs = (const float*)d_in[0];
  const int*   edges  = (const int*)d_in[1];
  const float* w_in   = (const float*)d_in[2];
  const float* b_in   = (const float*)d_in[3];
  const float* w1     = (const float*)d_in[4];
  const float* b1     = (const float*)d_in[5];
  const float* w2     = (const float*)d_in[6];
  const float* b2     = (const float*)d_in[7];
  const float* w_out  = (const float*)d_in[8];
  const float* b_out  = (const float*)d_in[9];

  const int H   = in_sizes[3];             // 128
  const int IN  = in_sizes[2] / H;         // 128
  const int N   = in_sizes[0] / IN;        // 50000
  const int E   = in_sizes[1] / 2;         // 800000
  const int OUT = in_sizes[9];             // 64
  const int B   = in_sizes[5] / H;         // 6

  const int* srcI = edges;
  const int* dstI = edges + E;

  float* norm = (float*)d_ws;
  float* S    = norm + (((size_t)N + 255) & ~(size_t)255);
  float* AGG  = S + (size_t)N * H;
  float* Hb   = AGG + (size_t)N * H;

  float* xout = (float*)d_out;              // output 0: [N, OUT]
  float* xbuf = xout + (size_t)N * OUT;     // output 1: [N, H] (doubles as x buffer)

  // degree -> norm = rsqrt(1 + deg)
  k_norm_init <<<(N + 255) / 256, 256, 0, stream>>>(norm, N);
  k_norm_count<<<(E + 255) / 256, 256, 0, stream>>>(dstI, norm, E);
  k_norm_rsqrt<<<(N + 255) / 256, 256, 0, stream>>>(norm, N);

  auto conv = [&](const float* Xin, const float* W, const float* bias, int K, int Hout,
                  float* outp, int mode, const float* resid) {
    size_t shmem = (size_t)(128 + Hout) * (K + 4) * sizeof(float);
    k_gemm_wmma<<<(N + 127) / 128, 256, shmem, stream>>>(Xin, W, norm, S, AGG, N, K, Hout);
    long tot = (long)E * (Hout >> 2);
    k_scatter<<<(unsigned)((tot + 255) / 256), 256, 0, stream>>>(srcI, dstI, S, AGG, E, Hout);
    long tot2 = (long)N * (Hout >> 2);
    k_finalize<<<(unsigned)((tot2 + 255) / 256), 256, 0, stream>>>(AGG, norm, bias, resid,
                                                                   outp, N, Hout, mode);
  };

  // x = leaky(conv(inputs))
  conv(inputs, w_in, b_in, IN, H, xbuf, 0, nullptr);
  // 6 bottleneck blocks
  for (int i = 0; i < B; ++i) {
    conv(xbuf, w1 + (size_t)i * H * H, b1 + (size_t)i * H, H, H, Hb,   0, nullptr);
    conv(Hb,   w2 + (size_t)i * H * H, b2 + (size_t)i * H, H, H, xbuf, 1, xbuf);
  }
  // x_out = conv(x) (no activation)
  conv(xbuf, w_out, b_out, H, OUT, xout, 2, nullptr);
}